// FlowAttention_51642686767977
// MI455X (gfx1250) — compile-verified
//
#include <hip/hip_runtime.h>
#include <hip/hip_bf16.h>

// FlowAttention for MI455X (gfx1250, wave32).
// Shapes: B=1, Dim=128, H=8, d=16, L = F*T = 262144.
//   K0: zero softmax denominators + kv accumulators (in ws).
//   K1: per-position stats. Sigmoided Q/K staged ONCE per position into LDS
//       (thread-private columns, conflict-free, no barriers), then 4 algebraic
//       passes read LDS. Softmax denominator per head via wave32 shfl_xor
//       reduction + atomicAdd (safe: conserved_source clipped to [-1,1]).
//   K2: kv[h] (16x16) = sum_l k^T * (w*v) via V_WMMA_F32_16X16X4_F32.
//       K/V tiles staged to LDS with gfx1250 async global->LDS copies
//       (ASYNCcnt); padded stride-17 layout keeps the WMMA feed conflict-free.
//   K3: per 16-position tile: qkv per head (4 WMMA k-steps) scaled by
//       sink_allocation into LDS x-tile; projection x @ W^T + b with
//       32 WMMA k-steps per 16-wide output tile.
// All reciprocals use v_rcp_f32 instead of IEEE-exact division sequences.

#define H 8
#define DH 16
#define DMODEL 128
#define EPS 1e-6f

typedef __attribute__((ext_vector_type(2))) float v2f;
typedef __attribute__((ext_vector_type(8))) float v8f;

#ifndef __has_builtin
#define __has_builtin(x) 0
#endif
#if __has_builtin(__builtin_amdgcn_global_load_async_to_lds_b32)
#define FA_ASYNC 1
#else
#define FA_ASYNC 0
#endif

typedef __attribute__((address_space(1))) int as1_int;
typedef __attribute__((address_space(3))) int as3_int;

__device__ __forceinline__ as1_int* as_global(const void* p) {
    return (as1_int*)(unsigned long long)p;
}
__device__ __forceinline__ as3_int* as_lds(void* p) {
    return (as3_int*)(unsigned int)(unsigned long long)p;
}

#if __has_builtin(__builtin_amdgcn_rcpf)
__device__ __forceinline__ float rcpf(float x) { return __builtin_amdgcn_rcpf(x); }
#else
__device__ __forceinline__ float rcpf(float x) { return 1.0f / x; }
#endif

__device__ __forceinline__ float sigf(float x) { return rcpf(1.0f + __expf(-x)); }

// ---------------- K0: zero accumulators ----------------
__global__ void fa_init(float* __restrict__ denom, float* __restrict__ kv) {
    int t = blockIdx.x * blockDim.x + threadIdx.x;
    if (t < H) denom[t] = 0.0f;
    if (t < H * DH * DH) kv[t] = 0.0f;
}

// ---------------- K1: per-position statistics ----------------
// 64 threads (2 waves), one position each; 64 KB LDS holds sigmoided Q/K.
__global__ __launch_bounds__(64) void fa_stats(
    const float* __restrict__ Q, const float* __restrict__ K,
    float* __restrict__ si_ws, float* __restrict__ sa_ws,
    float* __restrict__ es_ws, float* __restrict__ denom, int L) {
    const int t = threadIdx.x;
    const int l = blockIdx.x * 64 + t;

    __shared__ float sq[DMODEL * 64];   // [c][l-col]; each thread owns column t
    __shared__ float sk[DMODEL * 64];

    // Stage: coalesced global reads, sigmoid computed exactly once per element.
#pragma unroll
    for (int c = 0; c < DMODEL; ++c) {
        sq[c * 64 + t] = sigf(Q[(size_t)c * L + l]);
        sk[c * 64 + t] = sigf(K[(size_t)c * L + l]);
    }

    // Pass 1: head-sums qs[e], ks[e]
    float qs[DH], ks[DH];
#pragma unroll
    for (int e = 0; e < DH; ++e) { qs[e] = 0.0f; ks[e] = 0.0f; }
#pragma unroll
    for (int h = 0; h < H; ++h)
#pragma unroll
        for (int e = 0; e < DH; ++e) {
            qs[e] += sq[(h * DH + e) * 64 + t];
            ks[e] += sk[(h * DH + e) * 64 + t];
        }

    // Pass 2: sink_incoming (si), source_outgoing (so)
    float si[H], so[H];
#pragma unroll
    for (int h = 0; h < H; ++h) {
        float ai = 0.0f, ao = 0.0f;
#pragma unroll
        for (int e = 0; e < DH; ++e) {
            ai += (sq[(h * DH + e) * 64 + t] + EPS) * (ks[e] + EPS);
            ao += (sk[(h * DH + e) * 64 + t] + EPS) * (qs[e] + EPS);
        }
        si[h] = rcpf(ai);
        so[h] = rcpf(ao);
    }

    // Pass 3: kr[e] = sum_h k*so, qr[e] = sum_h q*si
    float kr[DH], qr[DH];
#pragma unroll
    for (int e = 0; e < DH; ++e) { kr[e] = 0.0f; qr[e] = 0.0f; }
#pragma unroll
    for (int h = 0; h < H; ++h)
#pragma unroll
        for (int e = 0; e < DH; ++e) {
            kr[e] += sk[(h * DH + e) * 64 + t] * so[h];
            qr[e] += sq[(h * DH + e) * 64 + t] * si[h];
        }

    // Pass 4: conserved sink/source -> allocation, exp(source); reduce exp sums
#pragma unroll
    for (int h = 0; h < H; ++h) {
        float cs = 0.0f, csr = 0.0f;
#pragma unroll
        for (int e = 0; e < DH; ++e) {
            cs  += (sq[(h * DH + e) * 64 + t] + EPS) * (kr[e] + EPS);
            csr += (sk[(h * DH + e) * 64 + t] + EPS) * (qr[e] + EPS);
        }
        csr = fminf(1.0f, fmaxf(-1.0f, csr));
        float es = __expf(csr);
        si_ws[(size_t)h * L + l] = si[h];
        sa_ws[(size_t)h * L + l] = sigf(cs);
        es_ws[(size_t)h * L + l] = es;

        // wave32 reduction of es, one atomicAdd per wave
        float v = es;
#pragma unroll
        for (int m = 16; m >= 1; m >>= 1) v += __shfl_xor(v, m, 32);
        if ((t & 31) == 0) atomicAdd(&denom[h], v);
    }
}

// ---------------- K2: kv accumulation via f32 WMMA ----------------
// Block: 256 threads = 8 waves. blockIdx.x = head, blockIdx.y = 256-position chunk.
__global__ __launch_bounds__(256) void fa_kv(
    const float* __restrict__ K, const float* __restrict__ V,
    const float* __restrict__ es_ws, const float* __restrict__ denom,
    float* __restrict__ kv, int L) {
    const int h = blockIdx.x;
    const int l0 = blockIdx.y * 256;
    const int t = threadIdx.x;

    __shared__ float sk[256 * 17];   // [l][e] raw K, stride 17 -> conflict-free
    __shared__ float sv[256 * 17];   // [l][e] raw V
    __shared__ float wl[256];
    __shared__ float red[8 * 256];

#if FA_ASYNC
    // gfx1250 async global->LDS copies: per-lane LDS destinations give the
    // transposed, padded layout directly; tracked by ASYNCcnt.
#pragma unroll
    for (int e = 0; e < DH; ++e) {
        __builtin_amdgcn_global_load_async_to_lds_b32(
            as_global(K + (size_t)(h * DH + e) * L + l0 + t),
            as_lds(&sk[t * 17 + e]), 0, 0);
        __builtin_amdgcn_global_load_async_to_lds_b32(
            as_global(V + (size_t)(h * DH + e) * L + l0 + t),
            as_lds(&sv[t * 17 + e]), 0, 0);
    }
#else
#pragma unroll
    for (int e = 0; e < DH; ++e) {
        sk[t * 17 + e] = K[(size_t)(h * DH + e) * L + l0 + t];
        sv[t * 17 + e] = V[(size_t)(h * DH + e) * L + l0 + t];
    }
#endif

    const float wscale = (float)L * rcpf(denom[h]);
    wl[t] = es_ws[(size_t)h * L + l0 + t] * wscale;

#if FA_ASYNC
#if __has_builtin(__builtin_amdgcn_s_wait_asynccnt)
    __builtin_amdgcn_s_wait_asynccnt(0);
#else
    asm volatile("s_wait_asynccnt 0x0" ::: "memory");
#endif
#endif
    __syncthreads();

    const int wave = t >> 5;
    const int lane = t & 31;
    const int em = lane & 15;            // row e (A) / col m (B)
    const int kb = (lane >> 4) << 1;     // K-slot base: 0 or 2

    v8f acc = {};
    const int lw = wave * 32;
#pragma unroll
    for (int s = 0; s < 8; ++s) {
        const int lb = lw + s * 4;
        v2f a, b;
        a.x = sigf(sk[(lb + kb) * 17 + em]) * wl[lb + kb];
        a.y = sigf(sk[(lb + kb + 1) * 17 + em]) * wl[lb + kb + 1];
        b.x = sv[(lb + kb) * 17 + em];
        b.y = sv[(lb + kb + 1) * 17 + em];
        acc = __builtin_amdgcn_wmma_f32_16x16x4_f32(false, a, false, b,
                                                    (short)0, acc, false, false);
    }

    // C layout: lanes 0-15 -> N=lane, M=r; lanes 16-31 -> N=lane-16, M=r+8.
    const int mb = (lane >> 4) * 8;
#pragma unroll
    for (int r = 0; r < 8; ++r)
        red[wave * 256 + (mb + r) * DH + em] = acc[r];
    __syncthreads();

    float s = 0.0f;
#pragma unroll
    for (int w2 = 0; w2 < 8; ++w2) s += red[w2 * 256 + t];
    atomicAdd(&kv[h * 256 + t], s);
}

// ---------------- K3: qkv + output projection via f32 WMMA ----------------
// Block: 256 threads = 8 waves, 16 positions per block.
__global__ __launch_bounds__(256) void fa_out(
    const float* __restrict__ Q, const float* __restrict__ si_ws,
    const float* __restrict__ sa_ws, const float* __restrict__ kv,
    const float* __restrict__ W, const float* __restrict__ bias,
    float* __restrict__ out, int L) {
    const int l0 = blockIdx.x * 16;
    const int t = threadIdx.x;
    const int wave = t >> 5;
    const int lane = t & 31;
    const int n = lane & 15;
    const int kb = (lane >> 4) << 1;
    const int mb = (lane >> 4) * 8;

    __shared__ float x_lds[16 * 130];    // [l][c], stride 130

    // Phase 1: wave = head. qkv = (q*si) @ kv_h, then * sink_allocation.
    {
        const int h = wave;
        const float sif = si_ws[(size_t)h * L + l0 + n];   // row M = lane&15
        v8f acc = {};
#pragma unroll
        for (int s = 0; s < 4; ++s) {
            const int e0 = 4 * s + kb;
            v2f a, b;
            a.x = sigf(Q[(size_t)(h * DH + e0) * L + l0 + n]) * sif;
            a.y = sigf(Q[(size_t)(h * DH + e0 + 1) * L + l0 + n]) * sif;
            b.x = kv[h * 256 + e0 * DH + n];
            b.y = kv[h * 256 + (e0 + 1) * DH + n];
            acc = __builtin_amdgcn_wmma_f32_16x16x4_f32(false, a, false, b,
                                                        (short)0, acc, false, false);
        }
#pragma unroll
        for (int r = 0; r < 8; ++r) {
            const int M = mb + r;
            x_lds[M * 130 + h * DH + n] =
                acc[r] * sa_ws[(size_t)h * L + l0 + M];
        }
    }
    __syncthreads();

    // Phase 2: wave = 16-wide output-column tile. out = x @ W^T + b.
    {
        const int co0 = wave * 16;
        const float bv = bias[co0 + n];
        v8f acc;
#pragma unroll
        for (int r = 0; r < 8; ++r) acc[r] = bv;
#pragma unroll
        for (int s = 0; s < 32; ++s) {
            const int ci = 4 * s + kb;
            v2f a, b;
            a.x = x_lds[n * 130 + ci];       // A row M = lane&15
            a.y = x_lds[n * 130 + ci + 1];
            b.x = W[(co0 + n) * DMODEL + ci];
            b.y = W[(co0 + n) * DMODEL + ci + 1];
            acc = __builtin_amdgcn_wmma_f32_16x16x4_f32(false, a, false, b,
                                                        (short)0, acc, false, false);
        }
#pragma unroll
        for (int r = 0; r < 8; ++r) {
            const int M = mb + r;
            out[(size_t)(co0 + n) * L + l0 + M] = acc[r];
        }
    }
}

// ---------------- launcher ----------------
extern "C" void kernel_launch(void* const* d_in, const int* in_sizes, int n_in,
                              void* d_out, int out_size, void* d_ws, size_t ws_size,
                              hipStream_t stream) {
    const float* Q = (const float*)d_in[0];
    const float* K = (const float*)d_in[1];
    const float* V = (const float*)d_in[2];
    const float* W = (const float*)d_in[3];
    const float* b = (const float*)d_in[4];
    float* out = (float*)d_out;
    const int L = in_sizes[0] / DMODEL;          // 262144

    float* ws = (float*)d_ws;
    float* si_ws = ws;
    float* sa_ws = ws + (size_t)H * L;
    float* es_ws = ws + (size_t)2 * H * L;
    float* denom = ws + (size_t)3 * H * L;
    float* kv    = denom + 16;

    fa_init<<<(H * DH * DH + 255) / 256, 256, 0, stream>>>(denom, kv);
    fa_stats<<<L / 64, 64, 0, stream>>>(Q, K, si_ws, sa_ws, es_ws, denom, L);
    fa_kv<<<dim3(H, L / 256), 256, 0, stream>>>(K, V, es_ws, denom, kv, L);
    fa_out<<<L / 16, 256, 0, stream>>>(Q, si_ws, sa_ws, kv, W, b, out, L);
}